// CoordinatorGNN_20066087207286
// MI455X (gfx1250) — compile-verified
//
#include <hip/hip_runtime.h>
#include <math.h>

typedef float v2f __attribute__((ext_vector_type(2)));
typedef float v8f __attribute__((ext_vector_type(8)));

#define HID 128
#define HEADS 4
#define HEAD_DIM 32
#define N_A 512
#define N_R 1024

// ---------------------------------------------------------------------------
// Small input projection with ReLU: out[n][j] = relu(sum_k x[n][k]W[k][j]+b[j])
// ---------------------------------------------------------------------------
__global__ void k_proj(const float* __restrict__ x, const float* __restrict__ W,
                       const float* __restrict__ b, float* __restrict__ out,
                       int n, int K) {
  int idx = blockIdx.x * blockDim.x + threadIdx.x;
  if (idx >= n * HID) return;
  int row = idx >> 7, col = idx & 127;
  float acc = b[col];
  const float* xr = x + row * K;
  for (int k = 0; k < K; ++k) acc += xr[k] * W[k * HID + col];
  out[idx] = acc > 0.f ? acc : 0.f;
}

// ---------------------------------------------------------------------------
// FP32 WMMA GEMM: C[N x 128] = X[N x 128] @ W[128 x 128] (+ bias)
// One wave (32 lanes) computes one 16x16 tile via V_WMMA_F32_16X16X4_F32.
// A layout: lanes 0-15 hold (M=lane, K=k0,k0+1); lanes 16-31 (M=lane-16, K=k0+2,k0+3)
// B layout mirrors A with N in place of M. C/D: VGPR j -> M = j + 8*(lane>=16).
// ---------------------------------------------------------------------------
__global__ __launch_bounds__(32)
void k_gemm_wmma_f32(const float* __restrict__ X, const float* __restrict__ W,
                     const float* __restrict__ bias, float* __restrict__ C) {
  int mt = blockIdx.x >> 3;
  int nt = blockIdx.x & 7;
  int lane = threadIdx.x;
  int half = lane >> 4;      // 0 or 1
  int l16  = lane & 15;
  int arow = mt * 16 + l16;  // A row carried by this lane
  int col  = nt * 16 + l16;  // B/C column carried by this lane
  const float* xp = X + arow * HID;

  v8f acc = {0.f, 0.f, 0.f, 0.f, 0.f, 0.f, 0.f, 0.f};
  for (int k0 = 0; k0 < HID; k0 += 4) {
    int ka = k0 + half * 2;
    v2f a, b;
    a[0] = xp[ka];
    a[1] = xp[ka + 1];
    b[0] = W[ka * HID + col];
    b[1] = W[(ka + 1) * HID + col];
    acc = __builtin_amdgcn_wmma_f32_16x16x4_f32(
        /*neg_a=*/false, a, /*neg_b=*/false, b,
        /*c_mod=*/(short)0, acc, /*reuse_a=*/false, /*reuse_b=*/false);
  }
  float bv = bias ? bias[col] : 0.f;
#pragma unroll
  for (int j = 0; j < 8; ++j) {
    C[(mt * 16 + half * 8 + j) * HID + col] = acc[j] + bv;
  }
}

// ---------------------------------------------------------------------------
// Fill buffer with constant
// ---------------------------------------------------------------------------
__global__ void k_fill(float* __restrict__ p, int n, float v) {
  int i = blockIdx.x * blockDim.x + threadIdx.x;
  if (i < n) p[i] = v;
}

// ---------------------------------------------------------------------------
// Edge helpers (GATv2): logit = att[h] . leaky_relu(xj[src]+xi[dst], 0.2)
// ---------------------------------------------------------------------------
__device__ __forceinline__ void edge_nodes(const int* __restrict__ esrc,
                                           const int* __restrict__ edst,
                                           int Ereal, int e, int& s, int& d) {
  if (e < Ereal) { s = esrc[e]; d = edst[e]; }
  else           { s = e - Ereal; d = e - Ereal; }   // self loops
}

__device__ __forceinline__ float edge_logit(const float* __restrict__ xj,
                                            const float* __restrict__ xi,
                                            const float* __restrict__ att,
                                            int s, int d, int h) {
  const float* pj = xj + s * HID + h * HEAD_DIM;
  const float* pi = xi + d * HID + h * HEAD_DIM;
  const float* pa = att + h * HEAD_DIM;
  float acc = 0.f;
#pragma unroll
  for (int k = 0; k < HEAD_DIM; ++k) {
    float v = pj[k] + pi[k];
    v = v > 0.f ? v : 0.2f * v;
    acc += v * pa[k];
  }
  return acc;
}

__device__ __forceinline__ void atomicMaxF(float* addr, float val) {
  int old = __float_as_int(*addr);
  while (__int_as_float(old) < val) {
    int assumed = old;
    old = atomicCAS((int*)addr, assumed, __float_as_int(val));
    if (old == assumed) break;
  }
}

// Pass 1: segment max of logits per (dst, head)
__global__ void k_edge_max(const float* __restrict__ xj, const float* __restrict__ xi,
                           const float* __restrict__ att,
                           const int* __restrict__ esrc, const int* __restrict__ edst,
                           int Ereal, int Etot, float* __restrict__ m) {
  int idx = blockIdx.x * blockDim.x + threadIdx.x;
  if (idx >= Etot * HEADS) return;
  int e = idx >> 2, h = idx & 3;
  int s, d; edge_nodes(esrc, edst, Ereal, e, s, d);
  float lg = edge_logit(xj, xi, att, s, d, h);
  atomicMaxF(&m[d * HEADS + h], lg);
}

// Pass 2: segment sum of exp(logit - max)
__global__ void k_edge_denom(const float* __restrict__ xj, const float* __restrict__ xi,
                             const float* __restrict__ att,
                             const int* __restrict__ esrc, const int* __restrict__ edst,
                             int Ereal, int Etot,
                             const float* __restrict__ m, float* __restrict__ denom) {
  int idx = blockIdx.x * blockDim.x + threadIdx.x;
  if (idx >= Etot * HEADS) return;
  int e = idx >> 2, h = idx & 3;
  int s, d; edge_nodes(esrc, edst, Ereal, e, s, d);
  float lg = edge_logit(xj, xi, att, s, d, h);
  float a = __expf(lg - m[d * HEADS + h]);
  atomicAdd(&denom[d * HEADS + h], a);
}

// Pass 3: scatter-add alpha-weighted source messages into accumulator
__global__ void k_edge_scatter(const float* __restrict__ xj, const float* __restrict__ xi,
                               const float* __restrict__ att,
                               const int* __restrict__ esrc, const int* __restrict__ edst,
                               int Ereal, int Etot,
                               const float* __restrict__ m, const float* __restrict__ denom,
                               float* __restrict__ out) {
  int idx = blockIdx.x * blockDim.x + threadIdx.x;
  if (idx >= Etot * HEADS) return;
  int e = idx >> 2, h = idx & 3;
  int s, d; edge_nodes(esrc, edst, Ereal, e, s, d);
  float lg = edge_logit(xj, xi, att, s, d, h);
  float alpha = __expf(lg - m[d * HEADS + h]) / denom[d * HEADS + h];
  const float* pj = xj + s * HID + h * HEAD_DIM;
  float* po = out + d * HID + h * HEAD_DIM;
#pragma unroll
  for (int k = 0; k < HEAD_DIM; ++k) atomicAdd(&po[k], pj[k] * alpha);
}

// h = relu(accum + bias_gat1 + bias_gat2)  (in place)
__global__ void k_relu_bias(float* __restrict__ h, const float* __restrict__ b1,
                            const float* __restrict__ b2, int n) {
  int idx = blockIdx.x * blockDim.x + threadIdx.x;
  if (idx >= n * HID) return;
  int j = idx & 127;
  float v = h[idx] + b1[j] + b2[j];
  h[idx] = v > 0.f ? v : 0.f;
}

// ---------------------------------------------------------------------------
// Score head: scores[a,r] = W2 . relu(ap[a] + rp[r] + b1) + b2
// 16x16 pair tile per block, ap/rp/b1/W2 staged in LDS.
// ---------------------------------------------------------------------------
__global__ __launch_bounds__(256)
void k_score(const float* __restrict__ ap, const float* __restrict__ rp,
             const float* __restrict__ b1, const float* __restrict__ W2,
             const float* __restrict__ b2, float* __restrict__ out) {
  __shared__ float apS[16 * HID];
  __shared__ float rpS[16 * HID];
  __shared__ float b1S[HID];
  __shared__ float w2S[HID];
  int tx = threadIdx.x, ty = threadIdx.y;
  int tid = ty * 16 + tx;
  int a0 = blockIdx.x * 16, r0 = blockIdx.y * 16;
  for (int i = tid; i < 16 * HID; i += 256) {
    apS[i] = ap[a0 * HID + i];
    rpS[i] = rp[r0 * HID + i];
  }
  if (tid < HID) { b1S[tid] = b1[tid]; w2S[tid] = W2[tid]; }
  __syncthreads();
  float acc = 0.f;
  const float* apr = apS + ty * HID;
  const float* rpr = rpS + tx * HID;
#pragma unroll 4
  for (int k = 0; k < HID; ++k) {
    float v = apr[k] + rpr[k] + b1S[k];
    v = v > 0.f ? v : 0.f;
    acc += v * w2S[k];
  }
  out[(a0 + ty) * N_R + (r0 + tx)] = acc + b2[0];
}

// ---------------------------------------------------------------------------
// Host orchestration
// ---------------------------------------------------------------------------
struct GatP {
  const float *Wl, *Wr, *att, *bias, *bl, *br;
};

static GatP get_gat(void* const* d_in, int base) {
  GatP p;
  p.Wl   = (const float*)d_in[base + 0];
  p.Wr   = (const float*)d_in[base + 1];
  p.att  = (const float*)d_in[base + 2];
  p.bias = (const float*)d_in[base + 3];
  p.bl   = (const float*)d_in[base + 4];
  p.br   = (const float*)d_in[base + 5];
  return p;
}

static void run_gat(hipStream_t stream,
                    const float* xsrc, int nsrc, const float* xdst, int ndst,
                    const int* esrc, const int* edst, int Ereal, int nloop,
                    const GatP& p, float* out_accum,
                    float* xj, float* xi, float* m, float* denom) {
  // lin_l(source), lin_r(target) — full fp32 WMMA GEMMs
  k_gemm_wmma_f32<<<(nsrc / 16) * 8, 32, 0, stream>>>(xsrc, p.Wl, p.bl, xj);
  k_gemm_wmma_f32<<<(ndst / 16) * 8, 32, 0, stream>>>(xdst, p.Wr, p.br, xi);
  int nm = ndst * HEADS;
  k_fill<<<(nm + 255) / 256, 256, 0, stream>>>(m, nm, -3.0e38f);
  k_fill<<<(nm + 255) / 256, 256, 0, stream>>>(denom, nm, 0.f);
  int Etot = Ereal + nloop;
  int nth = Etot * HEADS;
  int nb = (nth + 255) / 256;
  k_edge_max<<<nb, 256, 0, stream>>>(xj, xi, p.att, esrc, edst, Ereal, Etot, m);
  k_edge_denom<<<nb, 256, 0, stream>>>(xj, xi, p.att, esrc, edst, Ereal, Etot, m, denom);
  k_edge_scatter<<<nb, 256, 0, stream>>>(xj, xi, p.att, esrc, edst, Ereal, Etot, m, denom, out_accum);
}

extern "C" void kernel_launch(void* const* d_in, const int* in_sizes, int n_in,
                              void* d_out, int out_size, void* d_ws, size_t ws_size,
                              hipStream_t stream) {
  (void)in_sizes; (void)n_in; (void)out_size; (void)ws_size;

  const float* x_agent  = (const float*)d_in[0];
  const float* x_region = (const float*)d_in[1];
  const int* ar_src = (const int*)d_in[2];
  const int* ar_dst = (const int*)d_in[3];
  const int* ra_src = (const int*)d_in[4];
  const int* ra_dst = (const int*)d_in[5];
  const int* aa_src = (const int*)d_in[6];
  const int* aa_dst = (const int*)d_in[7];
  const int* rr_src = (const int*)d_in[8];
  const int* rr_dst = (const int*)d_in[9];

  // params flattened by jax pytree (dict keys sorted):
  // 10: agent_proj.W  11: agent_proj.b
  // 12 + 24*li: layer li -> aa{Wl,Wr,att,bias,bl,br} ar{...} ra{...} rr{...}
  // 84: region_proj.W  85: region_proj.b
  // 86: score.W1 [256x128]  87: b1  88: W2 [128]  89: b2 [1]
  const float* Wa  = (const float*)d_in[10];
  const float* ba  = (const float*)d_in[11];
  const float* Wrg = (const float*)d_in[84];
  const float* brg = (const float*)d_in[85];
  const float* W1  = (const float*)d_in[86];
  const float* b1  = (const float*)d_in[87];
  const float* W2  = (const float*)d_in[88];
  const float* b2  = (const float*)d_in[89];

  // workspace layout (floats)
  float* ws = (float*)d_ws;
  float* h_a  = ws;                       // 512*128
  float* h_r  = h_a  + N_A * HID;         // 1024*128
  float* h_a2 = h_r  + N_R * HID;         // 512*128
  float* h_r2 = h_a2 + N_A * HID;         // 1024*128
  float* xj   = h_r2 + N_R * HID;         // 1024*128 (max)
  float* xi   = xj   + N_R * HID;         // 1024*128 (max)
  float* m    = xi   + N_R * HID;         // 1024*4
  float* den  = m    + N_R * HEADS;       // 1024*4

  // input projections
  k_proj<<<(N_A * HID + 255) / 256, 256, 0, stream>>>(x_agent, Wa, ba, h_a, N_A, 24);
  k_proj<<<(N_R * HID + 255) / 256, 256, 0, stream>>>(x_region, Wrg, brg, h_r, N_R, 20);

  float* cur_a = h_a; float* cur_r = h_r;
  float* new_a = h_a2; float* new_r = h_r2;

  for (int li = 0; li < 3; ++li) {
    int base = 12 + li * 24;
    GatP p_aa = get_gat(d_in, base + 0);
    GatP p_ar = get_gat(d_in, base + 6);
    GatP p_ra = get_gat(d_in, base + 12);
    GatP p_rr = get_gat(d_in, base + 18);

    k_fill<<<(N_A * HID + 255) / 256, 256, 0, stream>>>(new_a, N_A * HID, 0.f);
    k_fill<<<(N_R * HID + 255) / 256, 256, 0, stream>>>(new_r, N_R * HID, 0.f);

    // agents <- regions (ra) + agents (aa with self loops)
    run_gat(stream, cur_r, N_R, cur_a, N_A, ra_src, ra_dst, 200000, 0,    p_ra, new_a, xj, xi, m, den);
    run_gat(stream, cur_a, N_A, cur_a, N_A, aa_src, aa_dst, 50000,  N_A,  p_aa, new_a, xj, xi, m, den);
    // regions <- agents (ar) + regions (rr with self loops)
    run_gat(stream, cur_a, N_A, cur_r, N_R, ar_src, ar_dst, 200000, 0,    p_ar, new_r, xj, xi, m, den);
    run_gat(stream, cur_r, N_R, cur_r, N_R, rr_src, rr_dst, 50000,  N_R,  p_rr, new_r, xj, xi, m, den);

    k_relu_bias<<<(N_A * HID + 255) / 256, 256, 0, stream>>>(new_a, p_ra.bias, p_aa.bias, N_A);
    k_relu_bias<<<(N_R * HID + 255) / 256, 256, 0, stream>>>(new_r, p_ar.bias, p_rr.bias, N_R);

    float* t;
    t = cur_a; cur_a = new_a; new_a = t;
    t = cur_r; cur_r = new_r; new_r = t;
  }

  // score head projections (no bias; W1 split by row blocks)
  k_gemm_wmma_f32<<<(N_A / 16) * 8, 32, 0, stream>>>(cur_a, W1,             nullptr, xj); // ap
  k_gemm_wmma_f32<<<(N_R / 16) * 8, 32, 0, stream>>>(cur_r, W1 + HID * HID, nullptr, xi); // rp

  k_score<<<dim3(N_A / 16, N_R / 16), dim3(16, 16), 0, stream>>>(
      xj, xi, b1, W2, b2, (float*)d_out);
}